// MeanLoss_78434692760202
// MI455X (gfx1250) — compile-verified
//
#include <hip/hip_runtime.h>
#include <math.h>

typedef float v2f __attribute__((ext_vector_type(2)));
typedef float v4f __attribute__((ext_vector_type(4)));
typedef float v8f __attribute__((ext_vector_type(8)));

// Wave32 (gfx1250). Each wave processes a 16-row x 64-col tile.
// Squared-diff row sums are produced on the diagonal of a WMMA Gram matrix:
// acc = D * D^T (+acc), D = 16x4 chunk of (A-B). Diagonal = sum of squares.
// The 16x4 A-operand VGPR layout equals the transposed 4x16 B-operand layout,
// so passing the same registers as A and B yields D*D^T directly.
__global__ void __launch_bounds__(256)
rownorm_wmma_kernel(const float* __restrict__ A, const float* __restrict__ B,
                    float* __restrict__ partial, int nRows, float invN) {
    const int lane         = threadIdx.x & 31;
    const int waveInBlock  = threadIdx.x >> 5;
    const int wavesPerBlk  = blockDim.x >> 5;
    const int globalWave   = blockIdx.x * wavesPerBlk + waveInBlock;
    const int totalWaves   = gridDim.x * wavesPerBlk;
    const int nTiles       = (nRows + 15) >> 4;

    const int r = lane & 15;   // row within 16-row tile
    const int h = lane >> 4;   // half selector: K-slots {0,1} vs {2,3}

    float local = 0.0f;

    for (int tile = globalWave; tile < nTiles; tile += totalWaves) {
        const int row = tile * 16 + r;
        v8f acc = {};

        // Wave-uniform branch: EXEC stays all-ones on both paths (WMMA-safe).
        if (tile * 16 + 16 <= nRows) {
            // Fast path: full tile, no masking -> pk_sub feeds WMMA directly.
            const v4f* ap = (const v4f*)(A + (long long)row * 64 + (long long)h * 4);
            const v4f* bp = (const v4f*)(B + (long long)row * 64 + (long long)h * 4);
#pragma unroll
            for (int i = 0; i < 8; ++i) {
                v4f a = ap[i * 2];       // advance 8 floats/iter per lane
                v4f b = bp[i * 2];
                v2f d0, d1;
                d0.x = a.x - b.x;  d0.y = a.y - b.y;
                d1.x = a.z - b.z;  d1.y = a.w - b.w;
                acc = __builtin_amdgcn_wmma_f32_16x16x4_f32(
                    false, d0, false, d0, (short)0, acc, false, false);
                acc = __builtin_amdgcn_wmma_f32_16x16x4_f32(
                    false, d1, false, d1, (short)0, acc, false, false);
            }
        } else {
            // Tail tile: zero invalid rows multiplicatively (no EXEC divergence).
            const float msk = (row < nRows) ? 1.0f : 0.0f;
            const long long rowC = (row < nRows) ? row : (nRows - 1);
            const v4f* ap = (const v4f*)(A + rowC * 64 + (long long)h * 4);
            const v4f* bp = (const v4f*)(B + rowC * 64 + (long long)h * 4);
#pragma unroll
            for (int i = 0; i < 8; ++i) {
                v4f a = ap[i * 2];
                v4f b = bp[i * 2];
                v2f d0, d1;
                d0.x = (a.x - b.x) * msk;  d0.y = (a.y - b.y) * msk;
                d1.x = (a.z - b.z) * msk;  d1.y = (a.w - b.w) * msk;
                acc = __builtin_amdgcn_wmma_f32_16x16x4_f32(
                    false, d0, false, d0, (short)0, acc, false, false);
                acc = __builtin_amdgcn_wmma_f32_16x16x4_f32(
                    false, d1, false, d1, (short)0, acc, false, false);
            }
        }

        // Diagonal extraction: row m<8 -> lane m, VGPR m; row m>=8 -> lane m+16, VGPR m-8.
        const int dix = (lane < 8) ? lane : ((lane >= 24) ? (lane - 24) : -1);
        float s =
            (dix == 0) ? acc[0] :
            (dix == 1) ? acc[1] :
            (dix == 2) ? acc[2] :
            (dix == 3) ? acc[3] :
            (dix == 4) ? acc[4] :
            (dix == 5) ? acc[5] :
            (dix == 6) ? acc[6] :
            (dix == 7) ? acc[7] : 0.0f;
        local += (dix >= 0) ? sqrtf(s) * invN : 0.0f;
    }

    __shared__ float red[256];
    red[threadIdx.x] = local;
    __syncthreads();
    for (int k = 128; k > 0; k >>= 1) {
        if ((int)threadIdx.x < k) red[threadIdx.x] += red[threadIdx.x + k];
        __syncthreads();
    }
    if (threadIdx.x == 0) partial[blockIdx.x] = red[0];
}

__global__ void __launch_bounds__(256)
final_reduce_kernel(const float* __restrict__ partial, float* __restrict__ out, int n) {
    __shared__ float red[256];
    float s = 0.0f;
    for (int i = threadIdx.x; i < n; i += blockDim.x) s += partial[i];
    red[threadIdx.x] = s;
    __syncthreads();
    for (int k = 128; k > 0; k >>= 1) {
        if ((int)threadIdx.x < k) red[threadIdx.x] += red[threadIdx.x + k];
        __syncthreads();
    }
    if (threadIdx.x == 0) out[0] = red[0];
}

extern "C" void kernel_launch(void* const* d_in, const int* in_sizes, int n_in,
                              void* d_out, int out_size, void* d_ws, size_t ws_size,
                              hipStream_t stream) {
    const float* A = (const float*)d_in[0];
    const float* B = (const float*)d_in[1];
    float* out     = (float*)d_out;
    float* partial = (float*)d_ws;

    const int D = 64;
    const int N = in_sizes[0] / D;

    int nBlocks = 2048;                                    // 16384 waves; ~7.6 tiles each
    const int maxBlocks = (int)(ws_size / sizeof(float));
    if (nBlocks > maxBlocks) nBlocks = maxBlocks;
    if (nBlocks < 1) nBlocks = 1;

    rownorm_wmma_kernel<<<nBlocks, 256, 0, stream>>>(A, B, partial, N,
                                                     1.0f / (float)N);
    final_reduce_kernel<<<1, 256, 0, stream>>>(partial, out, nBlocks);
}